// WaveletDict_18322330485204
// MI455X (gfx1250) — compile-verified
//
#include <hip/hip_runtime.h>

#define IMG 1024
#define HW_N (1024*1024)
#define SCALE_C 0.3333333333333333f
#define HALF_SCALE_C 0.1666666666666667f
#define TAU_C 0.05f

typedef float v2f __attribute__((ext_vector_type(2)));
typedef float v8f __attribute__((ext_vector_type(8)));

// pywt dec_lo coefficients for db1..db8, padded to 16
__constant__ float DBTAB[8][16] = {
  {0.7071067811865476f, 0.7071067811865476f, 0,0,0,0,0,0,0,0,0,0,0,0,0,0},
  {-0.12940952255092145f, 0.22414386804185735f, 0.836516303737469f, 0.48296291314469025f,
   0,0,0,0,0,0,0,0,0,0,0,0},
  {0.035226291882100656f, -0.08544127388224149f, -0.13501102001039084f, 0.4598775021193313f,
   0.8068915093133388f, 0.3326705529509569f, 0,0,0,0,0,0,0,0,0,0},
  {-0.010597401784997278f, 0.032883011666982945f, 0.030841381835986965f, -0.18703481171888114f,
   -0.02798376941698385f, 0.6308807679295904f, 0.7148465705525415f, 0.23037781330885523f,
   0,0,0,0,0,0,0,0},
  {0.003335725285001549f, -0.012580751999015526f, -0.006241490213011705f, 0.07757149384006515f,
   -0.03224486958502952f, -0.24229488706619015f, 0.13842814590110342f, 0.7243085284385744f,
   0.6038292697974729f, 0.160102397974125f, 0,0,0,0,0,0},
  {-0.00107730108499558f, 0.004777257511010651f, 0.0005538422009938016f, -0.031582039318031156f,
   0.02752286553001629f, 0.09750160558707936f, -0.12976686756709563f, -0.22626469396516913f,
   0.3152503517092432f, 0.7511339080215775f, 0.4946238903983854f, 0.11154074335008017f,
   0,0,0,0},
  {0.0003537138000010399f, -0.0018016407039998328f, 0.00042957797300470274f, 0.012550998556013784f,
   -0.01657454163101562f, -0.03802993693503463f, 0.0806126091510659f, 0.07130921926705004f,
   -0.22403618499416572f, -0.14390600392910627f, 0.4697822874053586f, 0.7291320908465551f,
   0.39653931948230575f, 0.07785205408506236f, 0,0},
  {-0.00011747678400228192f, 0.0006754494059985568f, -0.0003917403729959771f, -0.00487035299301066f,
   0.008746094047015655f, 0.013981027917015516f, -0.04408825393106472f, -0.01736930100202211f,
   0.128747426620186f, 0.00047248457399797254f, -0.2840155429624281f, -0.015829105256023893f,
   0.5853546836548691f, 0.6756307362980128f, 0.3128715909144659f, 0.05441584224308161f}
};

// dec_lo[j] = T[j]; dec_hi[j] = (-1)^(j+1) T[L-1-j]
__device__ __forceinline__ float dec_coef(int f, int L, int hi, int j) {
  if (j < 0 || j >= L) return 0.0f;
  if (!hi) return DBTAB[f][j];
  float b = DBTAB[f][L - 1 - j];
  return (j & 1) ? b : -b;
}
// rec_lo[j] = T[L-1-j]; rec_hi[j] = (-1)^j T[j]
__device__ __forceinline__ float rec_coef(int f, int L, int hi, int j) {
  if (j < 0 || j >= L) return 0.0f;
  if (!hi) return DBTAB[f][L - 1 - j];
  float b = DBTAB[f][j];
  return (j & 1) ? -b : b;
}

// ---------------- elementwise kernels ----------------
__global__ void k_zero(float* __restrict__ p, long n) {
  long i = (long)blockIdx.x * blockDim.x + threadIdx.x;
  long s = (long)gridDim.x * blockDim.x;
  for (; i < n; i += s) p[i] = 0.0f;
}
__global__ void k_copy(float* __restrict__ d, const float* __restrict__ s, long n) {
  long i = (long)blockIdx.x * blockDim.x + threadIdx.x;
  long st = (long)gridDim.x * blockDim.x;
  for (; i < n; i += st) d[i] = s[i];
}
// x_new = relu(y - SCALE*recon); z = 2*x_new - x_old; x = x_new
__global__ void k_xz(const float* __restrict__ y, const float* __restrict__ recon,
                     float* __restrict__ x, float* __restrict__ z, long n) {
  long i = (long)blockIdx.x * blockDim.x + threadIdx.x;
  long st = (long)gridDim.x * blockDim.x;
  for (; i < n; i += st) {
    float xo = x[i];
    float xn = fmaxf(y[i] - SCALE_C * recon[i], 0.0f);
    x[i] = xn;
    z[i] = 2.0f * xn - xo;
  }
}
// v_tail = clip(v_tail + (SCALE/2)*z, +-TAU)
__global__ void k_vtail(float* __restrict__ vt, const float* __restrict__ z, long n) {
  long i = (long)blockIdx.x * blockDim.x + threadIdx.x;
  long st = (long)gridDim.x * blockDim.x;
  for (; i < n; i += st) {
    float t = vt[i] + HALF_SCALE_C * z[i];
    vt[i] = fminf(fmaxf(t, -TAU_C), TAU_C);
  }
}

// ---------------- horizontal analysis: lo/hi[r][i] = sum_j dec[j] * src[r][2i+1-j] ----------------
__global__ void k_ana_h(const float* __restrict__ src, int srcPitch, int n, int K,
                        float* __restrict__ lo, float* __restrict__ hi, int f, int L) {
  int i = blockIdx.x * blockDim.x + threadIdx.x;
  int r = blockIdx.y;
  if (i >= K || r >= n) return;
  const float* row = src + (size_t)r * srcPitch;
  int base = 2 * i + 1;
  int j0 = base - (n - 1); if (j0 < 0) j0 = 0;
  int j1 = (base < L - 1) ? base : (L - 1);
  float alo = 0.0f, ahi = 0.0f;
  for (int j = j0; j <= j1; ++j) {
    float xv = row[base - j];
    float dl = DBTAB[f][j];
    float db = DBTAB[f][L - 1 - j];
    float dh = (j & 1) ? db : -db;
    alo += dl * xv;
    ahi += dh * xv;
  }
  lo[(size_t)r * K + i] = alo;
  hi[(size_t)r * K + i] = ahi;
}

// ---------------- horizontal synthesis: out[r][c] (+)= sum rec_lo[j]*tlo[r][m] + rec_hi[j]*thi[r][m],
// m = (c+L-2-j)/2, j same parity as c ----------------
__global__ void k_synth_h(const float* __restrict__ tlo, const float* __restrict__ thi, int tPitch,
                          int rows, int s, float* __restrict__ out, int outPitch,
                          int f, int L, int accum) {
  int c = blockIdx.x * blockDim.x + threadIdx.x;
  int r = blockIdx.y;
  int mOut = 2 * s + 2 - L;
  if (c >= mOut || r >= rows) return;
  const float* rl = tlo + (size_t)r * tPitch;
  const float* rh = thi + (size_t)r * tPitch;
  float acc = 0.0f;
  for (int j = (c & 1); j < L; j += 2) {
    int num = c + L - 2 - j;
    if (num < 0) continue;
    int m = num >> 1;
    if (m >= s) continue;
    float clo = DBTAB[f][L - 1 - j];
    float cb  = DBTAB[f][j];
    float chi = (j & 1) ? -cb : cb;
    acc += clo * rl[m] + chi * rh[m];
  }
  size_t idx = (size_t)r * outPitch + c;
  if (accum) out[idx] += acc; else out[idx] = acc;
}

// ---------------- vertical analysis via WMMA f32 16x16x4 ----------------
// out[p][i] = sum_j dec[j] * src[2p+1-j][i]  -> banded matmul, tile-invariant weights.
// mode 0: plain store; mode 1: dual update dst = clip(dst + (SCALE/2)*val, +-TAU)
__global__ void __launch_bounds__(256)
k_ana_v_wmma(const float* __restrict__ src, int srcPitch, int nRows, int width,
             float* __restrict__ dst, int dstPitch, int outRows,
             int f, int L, int selHi, int mode) {
  int lane = threadIdx.x & 31;
  int wave = threadIdx.x >> 5;
  int tilesX = (width + 15) >> 4;
  int tilesY = (outRows + 15) >> 4;
  int tile = blockIdx.x * (blockDim.x >> 5) + wave;
  if (tile >= tilesX * tilesY) return;            // wave-uniform
  int tp = tile / tilesX, ti = tile - tp * tilesX;
  int p0 = tp << 4, i0 = ti << 4;
  int mrow = lane & 15, half = lane >> 4;
  int col = i0 + mrow;
  bool colOK = (col < width);
  int rBase = 2 * p0 + 2 - L;
  int chunks = (L + 33) >> 2;                     // ceil((L+30)/4)
  v8f acc = {};
  for (int c = 0; c < chunks; ++c) {
    int tt0 = (c << 2) + (half << 1);
    v2f a, b;
    a.x = dec_coef(f, L, selHi, 2 * mrow + L - 1 - tt0);
    a.y = dec_coef(f, L, selHi, 2 * mrow + L - 2 - tt0);
    int r0 = rBase + tt0;
    int r1 = r0 + 1;
    b.x = (colOK && r0 >= 0 && r0 < nRows) ? src[(size_t)r0 * srcPitch + col] : 0.0f;
    b.y = (colOK && r1 >= 0 && r1 < nRows) ? src[(size_t)r1 * srcPitch + col] : 0.0f;
    acc = __builtin_amdgcn_wmma_f32_16x16x4_f32(false, a, false, b, (short)0, acc, false, false);
  }
  if (colOK) {
    for (int v = 0; v < 8; ++v) {
      int row = p0 + v + (half << 3);
      if (row < outRows) {
        size_t idx = (size_t)row * dstPitch + col;
        if (mode == 0) dst[idx] = acc[v];
        else {
          float t = dst[idx] + HALF_SCALE_C * acc[v];
          dst[idx] = fminf(fmaxf(t, -TAU_C), TAU_C);
        }
      }
    }
  }
}

// ---------------- vertical synthesis via WMMA f32 16x16x4 ----------------
// out[t][i] = sum_m rec_lo[t+L-2-2m]*lo[m][i] + rec_hi[t+L-2-2m]*hi[m][i]
__global__ void __launch_bounds__(256)
k_synth_v_wmma(const float* __restrict__ blo, int loPitch,
               const float* __restrict__ bhi, int hiPitch,
               int s, float* __restrict__ dst, int dstPitch, int outRows, int f, int L) {
  int lane = threadIdx.x & 31;
  int wave = threadIdx.x >> 5;
  int tilesX = (s + 15) >> 4;
  int tilesY = (outRows + 15) >> 4;
  int tile = blockIdx.x * (blockDim.x >> 5) + wave;
  if (tile >= tilesX * tilesY) return;            // wave-uniform
  int tp = tile / tilesX, ti = tile - tp * tilesX;
  int t0 = tp << 4, i0 = ti << 4;
  int mrow = lane & 15, half = lane >> 4;
  int col = i0 + mrow;
  bool colOK = (col < s);
  int mBase = (t0 >> 1) - 1;
  v8f acc = {};
  for (int band = 0; band < 2; ++band) {
    const float* src = band ? bhi : blo;
    int pitch = band ? hiPitch : loPitch;
    for (int c = 0; c < 4; ++c) {
      int mm0 = (c << 2) + (half << 1);
      v2f a, b;
      a.x = rec_coef(f, L, band, mrow + L - 2 * mm0);
      a.y = rec_coef(f, L, band, mrow + L - 2 * (mm0 + 1));
      int m0 = mBase + mm0;
      int m1 = m0 + 1;
      b.x = (colOK && m0 >= 0 && m0 < s) ? src[(size_t)m0 * pitch + col] : 0.0f;
      b.y = (colOK && m1 >= 0 && m1 < s) ? src[(size_t)m1 * pitch + col] : 0.0f;
      acc = __builtin_amdgcn_wmma_f32_16x16x4_f32(false, a, false, b, (short)0, acc, false, false);
    }
  }
  if (colOK) {
    for (int v = 0; v < 8; ++v) {
      int row = t0 + v + (half << 3);
      if (row < outRows) dst[(size_t)row * dstPitch + col] = acc[v];
    }
  }
}

// ---------------- host orchestration ----------------
static inline int lvlsz(int n, int L) { return (n + L - 3) / 2 + 1; }

static void launch_ana_v(hipStream_t st, const float* src, int sp, int nRows, int width,
                         float* dst, int dp, int outRows, int f, int L, int selHi, int mode) {
  int tiles = ((width + 15) / 16) * ((outRows + 15) / 16);
  int blocks = (tiles + 7) / 8;
  k_ana_v_wmma<<<blocks, 256, 0, st>>>(src, sp, nRows, width, dst, dp, outRows, f, L, selHi, mode);
}
static void launch_synth_v(hipStream_t st, const float* blo, int lp, const float* bhi, int hp,
                           int s, float* dst, int dp, int outRows, int f, int L) {
  int tiles = ((s + 15) / 16) * ((outRows + 15) / 16);
  int blocks = (tiles + 7) / 8;
  k_synth_v_wmma<<<blocks, 256, 0, st>>>(blo, lp, bhi, hp, s, dst, dp, outRows, f, L);
}

extern "C" void kernel_launch(void* const* d_in, const int* in_sizes, int n_in,
                              void* d_out, int out_size, void* d_ws, size_t ws_size,
                              hipStream_t stream) {
  (void)in_sizes; (void)n_in;
  const float* y = (const float*)d_in[0];
  float* out = (float*)d_out;
  float* ws = (float*)d_ws;

  int Lf[8], k1a[8], k2a[8], k3a[8], m2a[8], m3a[8];
  size_t off[9]; off[0] = 0;
  for (int f = 0; f < 8; ++f) {
    int L = 2 * (f + 1);
    int k1 = lvlsz(IMG, L), k2 = lvlsz(k1, L), k3 = lvlsz(k2, L);
    Lf[f] = L; k1a[f] = k1; k2a[f] = k2; k3a[f] = k3;
    m3a[f] = 2 * k3 + 2 - L; m2a[f] = 2 * k2 + 2 - L;
    size_t seg = (size_t)k3 * k3 + 3ull * ((size_t)k1 * k1 + (size_t)k2 * k2 + (size_t)k3 * k3);
    off[f + 1] = off[f] + seg;
  }
  const size_t TOTAL = off[8] + (size_t)HW_N;
  const size_t TCAP = (size_t)1024 * 528;
  const size_t PCAP = (size_t)528 * 528;

  float* v     = ws;
  float* x     = v + TOTAL;
  float* z     = x + HW_N;
  float* recon = z + HW_N;
  float* T0    = recon + HW_N;
  float* T1    = T0 + TCAP;
  float* P0    = T1 + TCAP;
  float* P1    = P0 + PCAP;
  size_t needFloats = (size_t)(P1 + PCAP - ws);
  if (ws_size < needFloats * sizeof(float)) return;
  if ((size_t)out_size < (size_t)HW_N) return;

  // zero v and x (contiguous)
  k_zero<<<2048, 256, 0, stream>>>(v, (long)(TOTAL + HW_N));

  const int EB = 2048;
  for (int it = 0; it < 10; ++it) {
    // recon = identity part of v
    k_copy<<<EB, 256, 0, stream>>>(recon, v + off[8], (long)HW_N);

    // ---- Psi: synthesis per filter, accumulate into recon ----
    for (int f = 0; f < 8; ++f) {
      int L = Lf[f], k1 = k1a[f], k2 = k2a[f], k3 = k3a[f], m2 = m2a[f], m3 = m3a[f];
      size_t o = off[f];
      size_t oB3 = o + (size_t)k3 * k3;
      size_t oB2 = oB3 + 3ull * (size_t)k3 * k3;
      size_t oB1 = oB2 + 3ull * (size_t)k2 * k2;
      // level 3: (ll3,lh3)->T0, (hl3,hh3)->T1, combine -> P0 (m3 x m3)
      launch_synth_v(stream, v + o, k3, v + oB3, k3, k3, T0, k3, m3, f, L);
      launch_synth_v(stream, v + oB3 + (size_t)k3 * k3, k3, v + oB3 + 2ull * (size_t)k3 * k3, k3,
                     k3, T1, k3, m3, f, L);
      k_synth_h<<<dim3((m3 + 127) / 128, m3), 128, 0, stream>>>(T0, T1, k3, m3, k3, P0, m3, f, L, 0);
      // level 2: lo = P0 cropped to k2 (pitch m3), bands2 -> P1 (m2 x m2)
      launch_synth_v(stream, P0, m3, v + oB2, k2, k2, T0, k2, m2, f, L);
      launch_synth_v(stream, v + oB2 + (size_t)k2 * k2, k2, v + oB2 + 2ull * (size_t)k2 * k2, k2,
                     k2, T1, k2, m2, f, L);
      k_synth_h<<<dim3((m2 + 127) / 128, m2), 128, 0, stream>>>(T0, T1, k2, m2, k2, P1, m2, f, L, 0);
      // level 1: lo = P1 cropped to k1 (pitch m2), bands1 -> accumulate into recon (1024x1024)
      launch_synth_v(stream, P1, m2, v + oB1, k1, k1, T0, k1, IMG, f, L);
      launch_synth_v(stream, v + oB1 + (size_t)k1 * k1, k1, v + oB1 + 2ull * (size_t)k1 * k1, k1,
                     k1, T1, k1, IMG, f, L);
      k_synth_h<<<dim3((IMG + 127) / 128, IMG), 128, 0, stream>>>(T0, T1, k1, IMG, k1, recon, IMG, f, L, 1);
    }

    // ---- primal step + reflection ----
    k_xz<<<EB, 256, 0, stream>>>(y, recon, x, z, (long)HW_N);

    // ---- PsiT: analysis of z per filter, fused dual clip-update into v ----
    for (int f = 0; f < 8; ++f) {
      int L = Lf[f], k1 = k1a[f], k2 = k2a[f], k3 = k3a[f];
      size_t o = off[f];
      size_t oB3 = o + (size_t)k3 * k3;
      size_t oB2 = oB3 + 3ull * (size_t)k3 * k3;
      size_t oB1 = oB2 + 3ull * (size_t)k2 * k2;
      // level 1
      k_ana_h<<<dim3((k1 + 127) / 128, IMG), 128, 0, stream>>>(z, IMG, IMG, k1, T0, T1, f, L);
      launch_ana_v(stream, T0, k1, IMG, k1, P0, k1, k1, f, L, 0, 0);                         // ll1
      launch_ana_v(stream, T0, k1, IMG, k1, v + oB1, k1, k1, f, L, 1, 1);                    // lh1
      launch_ana_v(stream, T1, k1, IMG, k1, v + oB1 + (size_t)k1 * k1, k1, k1, f, L, 0, 1);  // hl1
      launch_ana_v(stream, T1, k1, IMG, k1, v + oB1 + 2ull * (size_t)k1 * k1, k1, k1, f, L, 1, 1); // hh1
      // level 2
      k_ana_h<<<dim3((k2 + 127) / 128, k1), 128, 0, stream>>>(P0, k1, k1, k2, T0, T1, f, L);
      launch_ana_v(stream, T0, k2, k1, k2, P1, k2, k2, f, L, 0, 0);                          // ll2
      launch_ana_v(stream, T0, k2, k1, k2, v + oB2, k2, k2, f, L, 1, 1);                     // lh2
      launch_ana_v(stream, T1, k2, k1, k2, v + oB2 + (size_t)k2 * k2, k2, k2, f, L, 0, 1);   // hl2
      launch_ana_v(stream, T1, k2, k1, k2, v + oB2 + 2ull * (size_t)k2 * k2, k2, k2, f, L, 1, 1); // hh2
      // level 3
      k_ana_h<<<dim3((k3 + 127) / 128, k2), 128, 0, stream>>>(P1, k2, k2, k3, T0, T1, f, L);
      launch_ana_v(stream, T0, k3, k2, k3, v + o, k3, k3, f, L, 0, 1);                       // ll3 (stored in v)
      launch_ana_v(stream, T0, k3, k2, k3, v + oB3, k3, k3, f, L, 1, 1);                     // lh3
      launch_ana_v(stream, T1, k3, k2, k3, v + oB3 + (size_t)k3 * k3, k3, k3, f, L, 0, 1);   // hl3
      launch_ana_v(stream, T1, k3, k2, k3, v + oB3 + 2ull * (size_t)k3 * k3, k3, k3, f, L, 1, 1); // hh3
    }

    // ---- identity part of dual update ----
    k_vtail<<<EB, 256, 0, stream>>>(v + off[8], z, (long)HW_N);
  }

  k_copy<<<EB, 256, 0, stream>>>(out, x, (long)HW_N);
}